// Conv2d_Mine_10977936408723
// MI455X (gfx1250) — compile-verified
//
#include <hip/hip_runtime.h>

// 7x7 stride-1 pad-3 fp32 convolution for (64, 512, 512) on gfx1250.
// - V_WMMA_F32_16X16X4_F32 does the 49-tap reduction as banded-Toeplitz
//   matmuls (exact fp32 accumulate, matches the fp32 reference).
// - Input halo is staged global->LDS via the CDNA5 async-to-LDS path
//   (ASYNCcnt) when the toolchain exposes it; zero padding handled per-lane.

typedef __attribute__((ext_vector_type(2))) float v2f;
typedef __attribute__((ext_vector_type(8))) float v8f;

#if __has_builtin(__builtin_amdgcn_global_load_async_to_lds_b32)
#define ASYNC_LDS 1
typedef __attribute__((address_space(1))) int gas_int;  // global
typedef __attribute__((address_space(3))) int las_int;  // LDS
#else
#define ASYNC_LDS 0
#endif

namespace {
constexpr int HH = 512;
constexpr int WW = 512;
constexpr int KK = 7;
constexpr int PAD = 3;

// Block tile: 8 waves, each wave owns a 16x16 output tile, arranged 2 (h) x 4 (w).
constexpr int BLK_H = 32;
constexpr int BLK_W = 64;
constexpr int IN_ROWS = BLK_H + KK - 1;        // 38
constexpr int IN_COLS_LD = 72;                 // 70 live cols (BLK_W+6) + 2 zero, even stride
constexpr int WPAD = 40;                       // padded weight row: index t+16, t in [-15, 23]
}  // namespace

__global__ __launch_bounds__(256) void conv7x7_wmma_f32(
    const float* __restrict__ x, const float* __restrict__ wgt,
    float* __restrict__ out) {
  __shared__ float sIn[IN_ROWS * IN_COLS_LD];  // 38*72*4 = 10944 B
  __shared__ float sW[KK * WPAD];              // 7*40*4  =  1120 B

  const int tid = threadIdx.x;
  const int b = blockIdx.z;
  const int h_base = blockIdx.y * BLK_H;
  const int w_base = blockIdx.x * BLK_W;

  // --- Zero-padded weight table: sW[dh*40 + 16 + t] = k[dh][t], 0 outside [0,7)
  for (int i = tid; i < KK * WPAD; i += 256) sW[i] = 0.0f;
  __syncthreads();
  if (tid < KK * KK) {
    const int dh = tid / KK, t = tid % KK;
    sW[dh * WPAD + 16 + t] = wgt[tid];
  }

  // --- Stage input halo (rows h_base-3..+34, cols w_base-3..+66) into LDS.
  //     Valid lanes use GLOBAL_LOAD_ASYNC_TO_LDS_B32 (no VGPR round-trip);
  //     out-of-image / padding lanes write zero via LDS store.
  const float* xb = x + b * (HH * WW);
  for (int i = tid; i < IN_ROWS * IN_COLS_LD; i += 256) {
    const int r = i / IN_COLS_LD;
    const int c = i % IN_COLS_LD;
    const int gh = h_base + r - PAD;
    const int gw = w_base + c - PAD;
    const bool valid = (c < BLK_W + 2 * PAD) && ((unsigned)gh < (unsigned)HH) &&
                       ((unsigned)gw < (unsigned)WW);
#if ASYNC_LDS
    if (valid) {
      __builtin_amdgcn_global_load_async_to_lds_b32(
          (gas_int*)(xb + gh * WW + gw), (las_int*)&sIn[i],
          /*imm offset=*/0, /*cpol=*/0);
    } else {
      sIn[i] = 0.0f;
    }
#else
    float v = 0.0f;
    if (valid) v = xb[gh * WW + gw];
    sIn[i] = v;
#endif
  }
#if ASYNC_LDS
#if __has_builtin(__builtin_amdgcn_s_wait_asynccnt)
  __builtin_amdgcn_s_wait_asynccnt(0);
#else
  asm volatile("s_wait_asynccnt 0" ::: "memory");
#endif
#endif
  __syncthreads();

  const int lane = tid & 31;
  const int wave = tid >> 5;
  const int oh = (wave >> 2) << 4;      // 0 | 16
  const int ow = (wave & 3) << 4;       // 0 | 16 | 32 | 48
  const int m = lane & 15;              // A-matrix row / D-matrix N
  const int sub = (lane >> 4) << 1;     // K sub-slot: lanes 16-31 hold K=2,3
  const int wb = 16 + sub - m;          // base into padded weight row (>= 1)

  v8f acc = {};
#pragma unroll
  for (int dh = 0; dh < KK; ++dh) {
    const float* aRow = &sIn[(oh + m + dh) * IN_COLS_LD + ow + sub];
    const float* bRow = &sW[dh * WPAD + wb];
#pragma unroll
    for (int c0 = 0; c0 < 6; ++c0) {
      // A[16x4]: I[h+dh, 4*c0 + sub .. +1]  (8B-aligned -> ds_load_b64)
      v2f a = *(const v2f*)(aRow + 4 * c0);
      // B[4x16]: Toeplitz band T[c, n] = k[dh, c - n]
      v2f bb;
      bb.x = bRow[4 * c0];
      bb.y = bRow[4 * c0 + 1];
      acc = __builtin_amdgcn_wmma_f32_16x16x4_f32(
          /*neg_a=*/false, a, /*neg_b=*/false, bb,
          /*c_mod=*/(short)0, acc, /*reuse_a=*/false, /*reuse_b=*/false);
    }
  }

  // --- D 16x16 layout: VGPR v -> M = v + 8*(lane>=16), N = lane&15
  float* ob = out + b * (HH * WW);
  const int row0 = h_base + oh + ((lane >> 4) << 3);
  const int colg = w_base + ow + m;
#pragma unroll
  for (int v = 0; v < 8; ++v) {
    ob[(row0 + v) * WW + colg] = acc[v];
  }
}

extern "C" void kernel_launch(void* const* d_in, const int* in_sizes, int n_in,
                              void* d_out, int out_size, void* d_ws, size_t ws_size,
                              hipStream_t stream) {
  const float* x = (const float*)d_in[0];
  const float* w = (const float*)d_in[1];
  float* out = (float*)d_out;
  const int batch = in_sizes[0] / (HH * WW);  // 64
  dim3 grid(WW / BLK_W, HH / BLK_H, batch);   // (8, 16, 64)
  conv7x7_wmma_f32<<<grid, 256, 0, stream>>>(x, w, out);
}